// DynamicConv1D_65730179498325
// MI455X (gfx1250) — compile-verified
//
#include <hip/hip_runtime.h>

typedef __attribute__((ext_vector_type(2))) float v2f;
typedef __attribute__((ext_vector_type(8))) float v8f;

#define B_   8
#define T_   2048
#define D_   512
#define K_   3
#define F_   512
#define ROWS (B_ * T_)   // 16384 rows, divisible by 16; tiles never cross batch edges

// ---------------------------------------------------------------------------
// Phase 1: s[row] = sum_d x[row*D + d], 16 rows per wave via
// V_WMMA_F32_16X16X4_F32 with an all-ones B matrix (D[m,n] = rowsum(m)).
// A-matrix 32-bit 16x4 layout: lanes 0-15 hold K={0,1}, lanes 16-31 K={2,3},
// M = lane & 15 -> per-lane float2 load from row (r0 + m), col kk + 2*(lane>>4).
// ---------------------------------------------------------------------------
__global__ __launch_bounds__(256)
void dynconv_rowsum_wmma(const float* __restrict__ x, float* __restrict__ s) {
    const int wave = (blockIdx.x * blockDim.x + threadIdx.x) >> 5;
    const int lane = threadIdx.x & 31;
    const int r0   = wave * 16;
    const int m    = lane & 15;
    const int kh   = (lane >> 4) << 1;         // 0 or 2
    const float* rowp = x + (size_t)(r0 + m) * D_ + kh;

    v8f c = {0.f, 0.f, 0.f, 0.f, 0.f, 0.f, 0.f, 0.f};
    v2f ones; ones[0] = 1.0f; ones[1] = 1.0f;  // B = all ones -> layout-invariant

    #pragma unroll 4
    for (int kk = 0; kk < D_; kk += 4) {
        v2f a = *(const v2f*)(rowp + kk);      // global_load_b64
        // 8 args: (neg_a, A, neg_b, B, c_mod, C, reuse_a, reuse_b)
        c = __builtin_amdgcn_wmma_f32_16x16x4_f32(
                false, a, false, ones, (short)0, c, false, false);
    }

    // C/D 16x16 f32 layout: lanes 0-15 VGPR j = M=j ; lanes 16-31 VGPR j = M=8+j.
    if (lane == 0) {
        #pragma unroll
        for (int j = 0; j < 8; ++j) s[r0 + j] = c[j];
    } else if (lane == 16) {
        #pragma unroll
        for (int j = 0; j < 8; ++j) s[r0 + 8 + j] = c[j];
    }
}

// ---------------------------------------------------------------------------
// Phase 2: out[row, f] = s[row-1]*k[row,0,f] + s[row]*k[row,1,f] + s[row+1]*k[row,2,f]
// with zero-padding at t==0 / t==T-1 (padding never crosses batches since
// t = row & (T_-1)). One thread per float4 of output; exact grid, no tail.
// ---------------------------------------------------------------------------
__global__ __launch_bounds__(256)
void dynconv_apply(const float* __restrict__ kern, const float* __restrict__ s,
                   float* __restrict__ out) {
    const int idx = blockIdx.x * blockDim.x + threadIdx.x;   // 0 .. ROWS*128-1
    const int row = idx >> 7;                                // F_/4 = 128 vec4 per row
    const int f4  = idx & 127;
    const int t   = row & (T_ - 1);

    const float4* kp = (const float4*)kern + (size_t)row * (K_ * (F_ / 4)) + f4;

    const float sc = s[row];
    float4 k1 = kp[F_ / 4];                                  // k = 1 (center tap)
    float4 acc;
    acc.x = sc * k1.x; acc.y = sc * k1.y; acc.z = sc * k1.z; acc.w = sc * k1.w;

    if (t > 0) {
        const float sm = s[row - 1];
        float4 k0 = kp[0];
        acc.x += sm * k0.x; acc.y += sm * k0.y; acc.z += sm * k0.z; acc.w += sm * k0.w;
    }
    if (t < T_ - 1) {
        const float sp = s[row + 1];
        float4 k2 = kp[2 * (F_ / 4)];
        acc.x += sp * k2.x; acc.y += sp * k2.y; acc.z += sp * k2.z; acc.w += sp * k2.w;
    }
    ((float4*)out)[idx] = acc;                               // global_store_b128
}

// ---------------------------------------------------------------------------
extern "C" void kernel_launch(void* const* d_in, const int* in_sizes, int n_in,
                              void* d_out, int out_size, void* d_ws, size_t ws_size,
                              hipStream_t stream) {
    const float* x    = (const float*)d_in[0];   // [B, T, D] f32
    const float* kern = (const float*)d_in[1];   // [B, T, K, F] f32
    float* s = (float*)d_ws;                     // ROWS floats = 64 KB scratch

    // Phase 1: 16384 rows / 16 rows-per-wave = 1024 waves; 8 waves/block.
    dynconv_rowsum_wmma<<<ROWS / 16 / 8, 256, 0, stream>>>(x, s);

    // Phase 2: ROWS * (F_/4) = 2,097,152 threads; 256/block -> 8192 blocks.
    dynconv_apply<<<(ROWS * (F_ / 4)) / 256, 256, 0, stream>>>(kern, s, (float*)d_out);
}